// GraphConvolutionalTransformer_27049704030977
// MI455X (gfx1250) — compile-verified
//
#include <hip/hip_runtime.h>
#include <math.h>

#define B_ 64
#define N_ 512
#define D_ 128
#define FMAXF 3.402823466e38f

typedef __attribute__((ext_vector_type(2))) float v2f;
typedef __attribute__((ext_vector_type(8))) float v8f;

__device__ __forceinline__ float wave_max32(float v) {
#pragma unroll
  for (int off = 16; off > 0; off >>= 1) v = fmaxf(v, __shfl_xor(v, off, 32));
  return v;
}
__device__ __forceinline__ float wave_sum32(float v) {
#pragma unroll
  for (int off = 16; off > 0; off >>= 1) v += __shfl_xor(v, off, 32);
  return v;
}

// ---------------------------------------------------------------------------
// Batched GEMM, fp32 via V_WMMA_F32_16X16X4_F32.
// Each wave computes a 32x32 output tile as 2x2 fragments (4 independent
// WMMAs per K-step to cover the WMMA->WMMA RAW hazard).
// A is [M,K] row-major. If transB: B is [N,K] row-major (element(k,n)=B[n*K+k]),
// else B is [K,N] row-major. C is [M,N]. blockIdx.z = batch.
// ---------------------------------------------------------------------------
__global__ __launch_bounds__(256) void gemm_wmma_f32(
    float* __restrict__ C, const float* __restrict__ A, const float* __restrict__ Bm,
    int N, int K, long sA, long sB, long sC, int transB, float alpha,
    const float* __restrict__ bias, int relu)
{
  const int lane = threadIdx.x & 31;
  const int wave = threadIdx.x >> 5;
  const int tilesN = N >> 5;                    // 32-wide tiles
  const int tileIdx = blockIdx.x * 8 + wave;
  const int tm = tileIdx / tilesN;
  const int tn = tileIdx - tm * tilesN;
  const int lr = lane & 15;
  const int hi = lane >> 4;                     // 0: K pair {0,1}; 1: K pair {2,3}

  const float* Ab = A + (long)blockIdx.z * sA;
  const float* Bb = Bm + (long)blockIdx.z * sB;
  float*       Cb = C + (long)blockIdx.z * sC;

  // A 16x4 f32 fragment layout: lanes 0-15 rows M=0..15 holding K={0,1},
  // lanes 16-31 same rows holding K={2,3} -> contiguous float2 per lane.
  const float* a0 = Ab + (size_t)(tm * 32 + lr) * K + 2 * hi;
  const float* a1 = a0 + (size_t)16 * K;

  v8f c00 = {}, c01 = {}, c10 = {}, c11 = {};

  if (transB) {
    const float* b0 = Bb + (size_t)(tn * 32 + lr) * K + 2 * hi;
    const float* b1 = b0 + (size_t)16 * K;
#pragma unroll 4
    for (int k0 = 0; k0 < K; k0 += 4) {
      v2f av0 = *(const v2f*)(a0 + k0);
      v2f av1 = *(const v2f*)(a1 + k0);
      v2f bv0 = *(const v2f*)(b0 + k0);
      v2f bv1 = *(const v2f*)(b1 + k0);
      c00 = __builtin_amdgcn_wmma_f32_16x16x4_f32(false, av0, false, bv0, (short)0, c00, false, false);
      c01 = __builtin_amdgcn_wmma_f32_16x16x4_f32(false, av0, false, bv1, (short)0, c01, false, false);
      c10 = __builtin_amdgcn_wmma_f32_16x16x4_f32(false, av1, false, bv0, (short)0, c10, false, false);
      c11 = __builtin_amdgcn_wmma_f32_16x16x4_f32(false, av1, false, bv1, (short)0, c11, false, false);
    }
  } else {
    // B 4x16 fragment: lanes 0-15 col n hold K={0,1}; lanes 16-31 hold K={2,3}.
    const float* b0 = Bb + (size_t)(2 * hi) * N + (tn * 32 + lr);
    const float* b1 = b0 + 16;
#pragma unroll 4
    for (int k0 = 0; k0 < K; k0 += 4) {
      v2f av0 = *(const v2f*)(a0 + k0);
      v2f av1 = *(const v2f*)(a1 + k0);
      v2f bv0, bv1;
      bv0.x = b0[(size_t)k0 * N];
      bv0.y = b0[(size_t)(k0 + 1) * N];
      bv1.x = b1[(size_t)k0 * N];
      bv1.y = b1[(size_t)(k0 + 1) * N];
      c00 = __builtin_amdgcn_wmma_f32_16x16x4_f32(false, av0, false, bv0, (short)0, c00, false, false);
      c01 = __builtin_amdgcn_wmma_f32_16x16x4_f32(false, av0, false, bv1, (short)0, c01, false, false);
      c10 = __builtin_amdgcn_wmma_f32_16x16x4_f32(false, av1, false, bv0, (short)0, c10, false, false);
      c11 = __builtin_amdgcn_wmma_f32_16x16x4_f32(false, av1, false, bv1, (short)0, c11, false, false);
    }
  }

  // C/D layout: VGPR r -> M = r (lanes 0-15) or M = 8+r (lanes 16-31), N = lane&15.
  const int colA = tn * 32 + lr;
  const int colB = colA + 16;
  const float bb0 = bias ? bias[colA] : 0.0f;
  const float bb1 = bias ? bias[colB] : 0.0f;
#pragma unroll
  for (int r = 0; r < 8; ++r) {
    const int row0 = tm * 32 + r + 8 * hi;
    const int row1 = row0 + 16;
    float v00 = c00[r] * alpha + bb0;
    float v01 = c01[r] * alpha + bb1;
    float v10 = c10[r] * alpha + bb0;
    float v11 = c11[r] * alpha + bb1;
    if (relu) {
      v00 = fmaxf(v00, 0.0f); v01 = fmaxf(v01, 0.0f);
      v10 = fmaxf(v10, 0.0f); v11 = fmaxf(v11, 0.0f);
    }
    Cb[(size_t)row0 * N + colA] = v00;
    Cb[(size_t)row0 * N + colB] = v01;
    Cb[(size_t)row1 * N + colA] = v10;
    Cb[(size_t)row1 * N + colB] = v11;
  }
}

// Row-wise stable softmax over N_=512, applying key-mask and guide-mask
// (reference subtracts FMAX, not -inf). One wave per row, 16 cols per lane.
__global__ __launch_bounds__(256) void softmax_mask_kernel(
    float* __restrict__ S, const float* __restrict__ masks, const float* __restrict__ guide)
{
  const int lane = threadIdx.x & 31;
  const int wave = threadIdx.x >> 5;
  const long row = (long)blockIdx.x * 8 + wave;   // over B*N
  const int b = (int)(row >> 9);                  // row / N_
  float* srow = S + row * (long)N_;
  const float* grow = guide + row * (long)N_;
  const float* mrow = masks + (long)b * N_;

  float e[16];
  float mx = -FMAXF;
#pragma unroll
  for (int j = 0; j < 16; ++j) {
    const int c = lane + 32 * j;
    float s = srow[c];
    if (mrow[c] == 0.0f) s -= FMAXF;   // attention mask (key position)
    if (grow[c] == 0.0f) s -= FMAXF;   // guide mask
    e[j] = s;
    mx = fmaxf(mx, s);
  }
  mx = wave_max32(mx);
  float sum = 0.0f;
#pragma unroll
  for (int j = 0; j < 16; ++j) { e[j] = __expf(e[j] - mx); sum += e[j]; }
  sum = wave_sum32(sum);
  const float inv = 1.0f / sum;
#pragma unroll
  for (int j = 0; j < 16; ++j) srow[lane + 32 * j] = e[j] * inv;
}

// out = layer_norm(a + res) * g + b, one wave per row of D_=128.
__global__ __launch_bounds__(256) void add_layernorm_kernel(
    float* __restrict__ out, const float* __restrict__ a, const float* __restrict__ res,
    const float* __restrict__ g, const float* __restrict__ bt)
{
  const int lane = threadIdx.x & 31;
  const int wave = threadIdx.x >> 5;
  const long row = (long)blockIdx.x * 8 + wave;
  const float* ap = a + row * (long)D_;
  const float* rp = res + row * (long)D_;
  float x[4];
  float s = 0.0f;
#pragma unroll
  for (int j = 0; j < 4; ++j) { const int c = lane + 32 * j; x[j] = ap[c] + rp[c]; s += x[j]; }
  s = wave_sum32(s);
  const float m = s * (1.0f / D_);
  float vs = 0.0f;
#pragma unroll
  for (int j = 0; j < 4; ++j) { const float d = x[j] - m; vs += d * d; }
  vs = wave_sum32(vs);
  const float rs = rsqrtf(vs * (1.0f / D_) + 1e-12f);
  float* op = out + row * (long)D_;
#pragma unroll
  for (int j = 0; j < 4; ++j) {
    const int c = lane + 32 * j;
    op[c] = (x[j] - m) * rs * g[c] + bt[c];
  }
}

__global__ void mask_mul_kernel(float* __restrict__ out, const float* __restrict__ f,
                                const float* __restrict__ masks, long total)
{
  const long i = (long)blockIdx.x * blockDim.x + threadIdx.x;
  if (i < total) out[i] = f[i] * masks[i >> 7];   // i / D_
}

__global__ void transpose128_kernel(float* __restrict__ o, const float* __restrict__ in)
{
  const int r = blockIdx.x, c = threadIdx.x;     // 128 x 128
  o[(size_t)c * 128 + r] = in[(size_t)r * 128 + c];
}

extern "C" void kernel_launch(void* const* d_in, const int* in_sizes, int n_in,
                              void* d_out, int out_size, void* d_ws, size_t ws_size,
                              hipStream_t stream)
{
  (void)in_sizes; (void)n_in; (void)out_size; (void)ws_size;
  const float* features = (const float*)d_in[0];
  const float* masks    = (const float*)d_in[1];
  const float* guide    = (const float*)d_in[2];
  const float* prior    = (const float*)d_in[3];
  const float* Wq       = (const float*)d_in[4];
  const float* Wk       = (const float*)d_in[5];
  const float* Wv       = (const float*)d_in[6];
  const float* W1       = (const float*)d_in[7];
  const float* b1       = (const float*)d_in[8];
  const float* W2       = (const float*)d_in[9];
  const float* b2       = (const float*)d_in[10];
  const float* ln1g     = (const float*)d_in[11];
  const float* ln1b     = (const float*)d_in[12];
  const float* ln2g     = (const float*)d_in[13];
  const float* ln2b     = (const float*)d_in[14];

  const size_t ND  = (size_t)N_ * D_;
  const size_t BND = (size_t)B_ * ND;
  const size_t NN  = (size_t)N_ * N_;
  const size_t BNN = (size_t)B_ * NN;
  const size_t DD  = (size_t)D_ * D_;

  float* out_feat = (float*)d_out;
  float* attn_all = out_feat + BND;

  float* ws   = (float*)d_ws;
  float* x    = ws;            // masked features
  float* q    = x + BND;       // also reused as ffn2 output
  float* k    = q + BND;
  float* v    = k + BND;
  float* post = v + BND;
  float* h    = post + BND;    // attn@v raw, then ffn1 hidden
  float* feat = h + BND;       // next-layer features
  float* WqT  = feat + BND;
  float* WkT  = WqT + DD;
  float* WvT  = WkT + DD;
  float* W1T  = WvT + DD;
  float* W2T  = W1T + DD;

  const float scale = 1.0f / sqrtf((float)D_);
  const dim3 blk(256);

  auto launch_gemm = [&](float* C, const float* A, const float* Bm,
                         int M, int Nn, int K, long sA, long sB, long sC,
                         int transB, float alpha, const float* bias, int relu) {
    dim3 grid((unsigned)((M / 32) * (Nn / 32) / 8), 1, B_);
    gemm_wmma_f32<<<grid, blk, 0, stream>>>(C, A, Bm, Nn, K, sA, sB, sC,
                                            transB, alpha, bias, relu);
  };

  for (int i = 0; i < 3; ++i) {
    const float* curf = (i == 0) ? features : feat;
    mask_mul_kernel<<<(unsigned)(BND / 256), 256, 0, stream>>>(x, curf, masks, (long)BND);

    transpose128_kernel<<<128, 128, 0, stream>>>(WqT, Wq + (size_t)i * DD);
    transpose128_kernel<<<128, 128, 0, stream>>>(WkT, Wk + (size_t)i * DD);
    transpose128_kernel<<<128, 128, 0, stream>>>(WvT, Wv + (size_t)i * DD);
    transpose128_kernel<<<128, 128, 0, stream>>>(W1T, W1 + (size_t)i * DD);
    transpose128_kernel<<<128, 128, 0, stream>>>(W2T, W2 + (size_t)i * DD);

    float* attn = attn_all + (size_t)i * BNN;
    if (i == 0) {
      hipMemcpyAsync(attn, prior, BNN * sizeof(float), hipMemcpyDeviceToDevice, stream);
    } else {
      launch_gemm(q, x, WqT, N_, D_, D_, (long)ND, 0, (long)ND, 1, 1.0f, nullptr, 0);
      launch_gemm(k, x, WkT, N_, D_, D_, (long)ND, 0, (long)ND, 1, 1.0f, nullptr, 0);
      launch_gemm(attn, q, k, N_, N_, D_, (long)ND, (long)ND, (long)NN, 1, scale, nullptr, 0);
      softmax_mask_kernel<<<(unsigned)(B_ * N_ / 8), 256, 0, stream>>>(attn, masks, guide);
    }

    launch_gemm(v, x, WvT, N_, D_, D_, (long)ND, 0, (long)ND, 1, 1.0f, nullptr, 0);
    launch_gemm(h, attn, v, N_, D_, N_, (long)NN, (long)ND, (long)ND, 0, 1.0f, nullptr, 0);
    add_layernorm_kernel<<<(unsigned)(B_ * N_ / 8), 256, 0, stream>>>(
        post, h, x, ln1g + (size_t)i * D_, ln1b + (size_t)i * D_);

    launch_gemm(h, post, W1T, N_, D_, D_, (long)ND, 0, (long)ND, 1, 1.0f, b1 + (size_t)i * D_, 1);
    launch_gemm(q, h, W2T, N_, D_, D_, (long)ND, 0, (long)ND, 1, 1.0f, b2 + (size_t)i * D_, 0);
    add_layernorm_kernel<<<(unsigned)(B_ * N_ / 8), 256, 0, stream>>>(
        feat, q, post, ln2g + (size_t)i * D_, ln2b + (size_t)i * D_);
  }

  mask_mul_kernel<<<(unsigned)(BND / 256), 256, 0, stream>>>(out_feat, feat, masks, (long)BND);
}